// DeformConv2d_73194832658527
// MI455X (gfx1250) — compile-verified
//
#include <hip/hip_runtime.h>
#include <math.h>

// Problem constants (match reference)
#define B_    8
#define H_    128
#define W_    128
#define C_    192
#define G_    6
#define GC_   32
#define K2_   9
#define NTOK  (B_ * H_ * W_)   // 131072 tokens
#define NOFF  108              // G*K2*2 offset columns
#define NMASK 54               // G*K2 mask columns
#define NCPAD 176              // 108 + 54 = 162 padded up to 11 * 16
#define MT    4                // M tiles (of 16 rows) per wave

typedef __attribute__((ext_vector_type(2))) float v2f;
typedef __attribute__((ext_vector_type(8))) float v8f;

// ---------------------------------------------------------------------------
// GEMM: D[nrows x ncols] = A[nrows x 192] * W[192 x ncols] + bias[ncols]
// Weights are pre-TRANSPOSED on device (Wt[n*192 + k]) so each lane's B
// column is K-contiguous: one aligned global_load_b64 per fragment, base
// pointer loop-invariant.
// One wave owns a 64x16 output strip: 4 accumulators reuse each B fragment
// 4x and give 4-way WMMA ILP. (ncols/16) waves per block share a 64x192
// A-slab staged in LDS via float4 (global_load_b128 + ds_store_b128).
// Fragment layouts per CDNA5 ISA 7.12.2:
//   A 16x4 : lane&15 = M row; lanes 0-15 hold K=0,1, lanes 16-31 K=2,3
//   B 4x16 : lane&15 = N col; lanes 0-15 hold K=0,1, lanes 16-31 K=2,3
//   C/D    : VGPR r -> M=r (lanes 0-15) / M=r+8 (lanes 16-31), N = lane&15
// ---------------------------------------------------------------------------
__global__ void __launch_bounds__(384)
gemm192_wmma(const float* __restrict__ A, const float* __restrict__ Wt,
             const float* __restrict__ bias, float* __restrict__ D, int ncols) {
  __shared__ float As[16 * MT * C_];   // 48 KB

  const int lane = threadIdx.x & 31;
  const int wave = threadIdx.x >> 5;
  const int row0 = blockIdx.x * (16 * MT);

  // Cooperative float4 stage of the 64x192 A slab
  {
    const float4* A4  = (const float4*)(A + (size_t)row0 * C_);
    float4*       As4 = (float4*)As;
    for (int i = threadIdx.x; i < (16 * MT * C_) / 4; i += blockDim.x)
      As4[i] = A4[i];
  }
  __syncthreads();

  const int col0  = wave << 4;
  const int mn    = lane & 15;          // M row (A) / N col (B, C/D)
  const int koff  = (lane >> 4) << 1;   // 0 for lanes 0-15, 2 for lanes 16-31
  const int mbase = (lane >> 4) << 3;   // C/D row base: 0 or 8

  // Lane's K-contiguous B column (transposed weight)
  const float* wcol = Wt + (size_t)(col0 + mn) * C_;

  v8f acc[MT] = {};

#if __has_builtin(__builtin_amdgcn_wmma_f32_16x16x4_f32)
  #pragma unroll 2
  for (int k0 = 0; k0 < C_; k0 += 4) {
    const v2f b = *(const v2f*)(wcol + k0 + koff);        // global_load_b64
    #pragma unroll
    for (int t = 0; t < MT; ++t) {
      const v2f a = *(const v2f*)&As[(t * 16 + mn) * C_ + k0 + koff];  // ds b64
      // 8 args: (neg_a, A, neg_b, B, c_mod, C, reuse_a, reuse_b)
      acc[t] = __builtin_amdgcn_wmma_f32_16x16x4_f32(
          false, a, false, b, (short)0, acc[t], false, false);
    }
  }
#else
  // Scalar fallback (layout-equivalent) — only taken if the builtin is absent.
  for (int t = 0; t < MT; ++t)
    for (int r = 0; r < 8; ++r) {
      float s = 0.f;
      for (int k = 0; k < C_; ++k)
        s += As[(t * 16 + mbase + r) * C_ + k] * wcol[k];
      acc[t][r] = s;
    }
#endif

  const float bv = bias[col0 + mn];
  #pragma unroll
  for (int t = 0; t < MT; ++t)
    #pragma unroll
    for (int r = 0; r < 8; ++r)
      D[(size_t)(row0 + t * 16 + mbase + r) * ncols + col0 + mn] = acc[t][r] + bv;
}

// ---------------------------------------------------------------------------
// Transpose a 192x192 weight: Wt[n*192 + k] = Wsrc[k*192 + n]
// ---------------------------------------------------------------------------
__global__ void __launch_bounds__(256)
transpose192(const float* __restrict__ Wsrc, float* __restrict__ Wt) {
  const int idx = blockIdx.x * blockDim.x + threadIdx.x;
  if (idx >= C_ * C_) return;
  const int n = idx / C_;
  const int k = idx - n * C_;
  Wt[idx] = Wsrc[k * C_ + n];
}

// ---------------------------------------------------------------------------
// Build the padded offset+mask weight directly TRANSPOSED:
//   wpad_t[n*192 + k], n in [0,176): cols 0..107 = w_off, 108..161 = w_mask,
//   162..175 = 0. Padded bias alongside.
// ---------------------------------------------------------------------------
__global__ void __launch_bounds__(256)
pack_offmask_t(const float* __restrict__ w_off, const float* __restrict__ b_off,
               const float* __restrict__ w_mask, const float* __restrict__ b_mask,
               float* __restrict__ wpad_t, float* __restrict__ bpad) {
  const int idx = blockIdx.x * blockDim.x + threadIdx.x;
  if (idx < NCPAD * C_) {
    const int n = idx / C_;
    const int k = idx - n * C_;
    float v = 0.f;
    if (n < NOFF)              v = w_off[k * NOFF + n];
    else if (n < NOFF + NMASK) v = w_mask[k * NMASK + (n - NOFF)];
    wpad_t[idx] = v;
  } else if (idx < NCPAD * C_ + NCPAD) {
    const int n = idx - NCPAD * C_;
    float v = 0.f;
    if (n < NOFF)              v = b_off[n];
    else if (n < NOFF + NMASK) v = b_mask[n - NOFF];
    bpad[n] = v;
  }
}

// ---------------------------------------------------------------------------
// Depthwise 3x3 (SAME, zero pad) + bias + SiLU. Memory bound, one thread per
// (b,h,w,c). w_dw is HWIO (3,3,1,C): w_dw[((kh*3)+kw)*C + c].
// ---------------------------------------------------------------------------
__global__ void __launch_bounds__(256)
dwconv3x3_silu(const float* __restrict__ x, const float* __restrict__ wdw,
               const float* __restrict__ bdw, float* __restrict__ out) {
  const size_t idx = (size_t)blockIdx.x * blockDim.x + threadIdx.x;
  if (idx >= (size_t)NTOK * C_) return;
  const int c  = (int)(idx % C_);
  const int t  = (int)(idx / C_);
  const int b  = t / (H_ * W_);
  const int hw = t - b * (H_ * W_);
  const int h  = hw / W_;
  const int w  = hw - h * W_;

  float s = bdw[c];
  #pragma unroll
  for (int dh = -1; dh <= 1; ++dh) {
    const int hh = h + dh;
    if ((unsigned)hh >= (unsigned)H_) continue;
    #pragma unroll
    for (int dw = -1; dw <= 1; ++dw) {
      const int ww = w + dw;
      if ((unsigned)ww >= (unsigned)W_) continue;
      s += x[(((size_t)b * H_ + hh) * W_ + ww) * C_ + c]
         * wdw[((dh + 1) * 3 + (dw + 1)) * C_ + c];
    }
  }
  out[idx] = s * (1.f / (1.f + __expf(-s)));  // SiLU
}

// ---------------------------------------------------------------------------
// Deformable sampling. One wave per (token, group); 32 lanes = 32 group
// channels (GC=32 matches wave32 exactly). Per-lane redundant 9-way softmax
// (broadcast loads), then 9 bilinear taps from x_proj with clip/trunc
// semantics matching the reference.
// ---------------------------------------------------------------------------
__global__ void __launch_bounds__(256)
deform_sample(const float* __restrict__ x_proj, const float* __restrict__ om_all,
              float* __restrict__ out) {
  const int wid  = blockIdx.x * (blockDim.x >> 5) + (threadIdx.x >> 5);
  const int lane = threadIdx.x & 31;
  if (wid >= NTOK * G_) return;

  const int g  = wid % G_;
  const int t  = wid / G_;
  const int b  = t / (H_ * W_);
  const int hw = t - b * (H_ * W_);
  const int h  = hw / W_;
  const int w  = hw - h * W_;

  const float* om = om_all + (size_t)t * NCPAD;

  // softmax over the 9 mask logits for this group
  float ek[K2_];
  float mx = -1e30f;
  #pragma unroll
  for (int k = 0; k < K2_; ++k) {
    ek[k] = om[NOFF + g * K2_ + k];
    mx = fmaxf(mx, ek[k]);
  }
  float den = 0.f;
  #pragma unroll
  for (int k = 0; k < K2_; ++k) { ek[k] = __expf(ek[k] - mx); den += ek[k]; }
  const float inv = 1.f / den;

  const float* base = x_proj + (size_t)b * (H_ * W_) * C_ + g * GC_ + lane;

  float acc = 0.f;
  #pragma unroll
  for (int k = 0; k < K2_; ++k) {
    const float oh = om[(g * K2_ + k) * 2 + 0] * 0.1f;  // OFFSET_SCALE
    const float ow = om[(g * K2_ + k) * 2 + 1] * 0.1f;
    float ph = (float)(k / 3 - 1) + oh + (float)h;
    float pw = (float)(k % 3 - 1) + ow + (float)w;
    ph = fminf(fmaxf(ph, 0.f), (float)(H_ - 1));
    pw = fminf(fmaxf(pw, 0.f), (float)(W_ - 1));
    const int hf = (int)ph;  // ph >= 0 -> trunc == floor (matches astype(int32))
    const int wf = (int)pw;
    const int hc = (hf + 1 < H_) ? hf + 1 : H_ - 1;
    const int wc = (wf + 1 < W_) ? wf + 1 : W_ - 1;
    const float fh = ph - (float)hf;
    const float fw = pw - (float)wf;
    const float v00 = base[(size_t)(hf * W_ + wf) * C_];
    const float v01 = base[(size_t)(hf * W_ + wc) * C_];
    const float v10 = base[(size_t)(hc * W_ + wf) * C_];
    const float v11 = base[(size_t)(hc * W_ + wc) * C_];
    const float sv = v00 * (1.f - fh) * (1.f - fw) + v01 * (1.f - fh) * fw
                   + v10 * fh * (1.f - fw)        + v11 * fh * fw;
    acc += sv * ek[k] * inv;
  }
  out[(size_t)t * C_ + g * GC_ + lane] = acc;
}

// ---------------------------------------------------------------------------
// Launch: pipeline on one stream; intermediates live in d_ws and stay
// L2-resident (each buffer ~100 MB; MI455X L2 = 192 MB) between producer and
// consumer.
//   ws layout (floats):
//     [x_proj | bufA | bufB | wt_in | wt_pw | wt_out | wpad_t | bpad]
//   bufA: xd_silu -> offmask(176 cols);  bufB: xd -> sampled
// ---------------------------------------------------------------------------
extern "C" void kernel_launch(void* const* d_in, const int* in_sizes, int n_in,
                              void* d_out, int out_size, void* d_ws, size_t ws_size,
                              hipStream_t stream) {
  const float* x      = (const float*)d_in[0];
  const float* w_in   = (const float*)d_in[1];
  const float* b_in   = (const float*)d_in[2];
  const float* w_dw   = (const float*)d_in[3];
  const float* b_dw   = (const float*)d_in[4];
  const float* w_pw   = (const float*)d_in[5];   // (1,1,C,C) == row-major CxC
  const float* b_pw   = (const float*)d_in[6];
  const float* w_off  = (const float*)d_in[7];
  const float* b_off  = (const float*)d_in[8];
  const float* w_mask = (const float*)d_in[9];
  const float* b_mask = (const float*)d_in[10];
  const float* w_out  = (const float*)d_in[11];
  const float* b_out  = (const float*)d_in[12];

  float* ws = (float*)d_ws;
  const size_t NTC = (size_t)NTOK * C_;
  float* x_proj = ws;                             // [NTOK x 192]
  float* bufA   = ws + NTC;                       // xd_silu, then offmask
  float* bufB   = ws + 2 * NTC;                   // xd, then sampled
  float* wt_in  = ws + 3 * NTC;                   // [192 x 192] transposed
  float* wt_pw  = wt_in + (size_t)C_ * C_;
  float* wt_out = wt_pw + (size_t)C_ * C_;
  float* wpad_t = wt_out + (size_t)C_ * C_;       // [176 x 192] transposed
  float* bpad   = wpad_t + (size_t)NCPAD * C_;    // [176]

  const int rowTiles = NTOK / (16 * MT);  // 2048 blocks, 64 output rows each
  const int wxw      = C_ * C_;           // 36864 weight elements

  // 0) transpose the three dense weights; build packed+transposed off/mask
  transpose192<<<(wxw + 255) / 256, 256, 0, stream>>>(w_in, wt_in);
  transpose192<<<(wxw + 255) / 256, 256, 0, stream>>>(w_pw, wt_pw);
  transpose192<<<(wxw + 255) / 256, 256, 0, stream>>>(w_out, wt_out);
  {
    const int n = NCPAD * C_ + NCPAD;
    pack_offmask_t<<<(n + 255) / 256, 256, 0, stream>>>(w_off, b_off, w_mask,
                                                        b_mask, wpad_t, bpad);
  }

  // 1) x_proj = x @ w_in + b_in           (WMMA f32)
  gemm192_wmma<<<rowTiles, (C_ / 16) * 32, 0, stream>>>(x, wt_in, b_in, x_proj, C_);

  // 2) depthwise 3x3 + SiLU -> bufA
  {
    const size_t n = NTC;
    dwconv3x3_silu<<<(unsigned)((n + 255) / 256), 256, 0, stream>>>(x, w_dw, b_dw, bufA);
  }

  // 3) xd = bufA @ w_pw + b_pw -> bufB    (WMMA f32)
  gemm192_wmma<<<rowTiles, (C_ / 16) * 32, 0, stream>>>(bufA, wt_pw, b_pw, bufB, C_);

  // 4) offmask = xd @ wpad + bpad -> bufA (WMMA f32, one fused GEMM)
  gemm192_wmma<<<rowTiles, (NCPAD / 16) * 32, 0, stream>>>(bufB, wpad_t, bpad, bufA, NCPAD);

  // 5) deformable bilinear sampling -> bufB
  {
    const int waves = NTOK * G_;              // 786432 waves, 8 per block
    deform_sample<<<(waves + 7) / 8, 256, 0, stream>>>(x_proj, bufA, bufB);
  }

  // 6) out = sampled @ w_out + b_out      (WMMA f32)
  gemm192_wmma<<<rowTiles, (C_ / 16) * 32, 0, stream>>>(bufB, wt_out, b_out, (float*)d_out, C_);
}